// TensorSnake_11235634446889
// MI455X (gfx1250) — compile-verified
//
#include <hip/hip_runtime.h>
#include <hip/hip_bf16.h>

// TensorSnake step, MI455X (gfx1250, wave32).
// One wave per game (32 lanes x 8 cells = 256 board cells), 8 waves / 256-thread block.
// Memory-bound: ~3KB per game streamed once -> NT hints everywhere.
// Cross-lane logic via ds_swizzle / v_readlane (wave32-native), no LDS barriers needed.
// rand_food tile prefetched via gfx1250 async global->LDS DMA (ASYNCcnt).

typedef int   v4i __attribute__((ext_vector_type(4)));
typedef float v4f __attribute__((ext_vector_type(4)));

#if __has_builtin(__builtin_amdgcn_global_load_async_to_lds_b128)
#define HAVE_ASYNC_LDS 1
typedef __attribute__((address_space(1))) v4i* global_v4i_ptr;
typedef __attribute__((address_space(3))) v4i* lds_v4i_ptr;
#else
#define HAVE_ASYNC_LDS 0
#endif

// Select c[sub] for wave-uniform sub without spilling the array to scratch.
__device__ __forceinline__ int sel8(const int c[8], int sub) {
    int v = c[0];
#pragma unroll
    for (int i = 1; i < 8; ++i) v = (sub == i) ? c[i] : v;
    return v;
}

// One butterfly step of the (value, index) argmax reduction over the wave.
// Tie-break: smaller linear index wins (matches jnp.argmax first-occurrence).
template <int IMM>
__device__ __forceinline__ void red_step(float& bv, int& bi) {
    int   ov_i = __builtin_amdgcn_ds_swizzle(__float_as_int(bv), IMM);
    int   oi   = __builtin_amdgcn_ds_swizzle(bi, IMM);
    float ov   = __int_as_float(ov_i);
    if ((ov > bv) || ((ov == bv) && (oi < bi))) { bv = ov; bi = oi; }
}

__global__ __launch_bounds__(256) void snake_step_kernel(
    const int*   __restrict__ action,
    const int*   __restrict__ pos_prev,
    const int*   __restrict__ pos_cur,
    const int*   __restrict__ state,
    const float* __restrict__ rand_food,
    float*       __restrict__ out,
    int G)
{
    const int lane = threadIdx.x & 31;
    const int wave = threadIdx.x >> 5;
    const int g    = blockIdx.x * 8 + wave;
    if (g >= G) return;                    // wave-uniform exit

    const size_t gbase = (size_t)g * 256;

#if HAVE_ASYNC_LDS
    // Prefetch this wave's rand_food tile into LDS asynchronously (ASYNCcnt);
    // it is only consumed at the argmax, well after the movement logic.
    __shared__ float lds_rand[8 * 256];
    {
        const float* gp = rand_food + gbase + (size_t)(lane * 8);
        float*       lp = &lds_rand[wave * 256 + lane * 8];
        __builtin_amdgcn_global_load_async_to_lds_b128(
            (global_v4i_ptr)gp, (lds_v4i_ptr)lp, 0, 0);
        __builtin_amdgcn_global_load_async_to_lds_b128(
            (global_v4i_ptr)(gp + 4), (lds_v4i_ptr)(lp + 4), 0, 0);
    }
#endif

    // ---- board: 8 cells per lane, 32B contiguous per lane, 1KB per wave ----
    const int* sp = state + gbase + lane * 8;
    v4i s0 = __builtin_nontemporal_load((const v4i*)sp);
    v4i s1 = __builtin_nontemporal_load((const v4i*)(sp + 4));
    int cells[8] = { s0.x, s0.y, s0.z, s0.w, s1.x, s1.y, s1.z, s1.w };

    // ---- wave-uniform per-game scalars (scalarized for SALU control flow) ----
    const int act = __builtin_amdgcn_readfirstlane(action[g]);
    int       pc0 = __builtin_amdgcn_readfirstlane(pos_cur[2 * g + 0]);
    int       pc1 = __builtin_amdgcn_readfirstlane(pos_cur[2 * g + 1]);
    const int pp0 = __builtin_amdgcn_readfirstlane(pos_prev[2 * g + 0]);
    const int pp1 = __builtin_amdgcn_readfirstlane(pos_prev[2 * g + 1]);

    const int dx = pc0 - pp0, dy = pc1 - pp1;
    int ndx = dx, ndy = dy;
    if (act == 0) { ndx = -dy; ndy =  dx; }   // turn left
    if (act == 2) { ndx =  dy; ndy = -dx; }   // turn right
    int nx = pc0 + ndx, ny = pc1 + ndy;
    const bool outside = (nx < 0) | (nx > 15) | (ny < 0) | (ny > 15);
    nx = nx < 0 ? 0 : (nx > 15 ? 15 : nx);
    ny = ny < 0 ? 0 : (ny > 15 ? 15 : ny);
    int next_lin = nx * 16 + ny;

    // Probe the target cell from the ORIGINAL board (broadcast via readlane).
    const int  cell    = __builtin_amdgcn_readlane(sel8(cells, next_lin & 7),
                                                   next_lin >> 3);
    const bool dead    = outside || (cell > 0);
    const bool feeding = (cell == -1);

    if (dead) {   // reset: (8,6)=1 (8,7)=2 (8,8)=1 -> linear 134,135,136
#pragma unroll
        for (int i = 0; i < 8; ++i) cells[i] = 0;
        if (lane == 16) { cells[6] = 1; cells[7] = 2; }
        if (lane == 17) { cells[0] = 1; }
        pc0 = 8; pc1 = 7;
        next_lin = 8 * 16 + 8;
    }
    const int  cur_lin = pc0 * 16 + pc1;
    const bool spawn   = dead || feeding;

    // ---- food argmax over empty cells (post-reset board) ----
#if HAVE_ASYNC_LDS
#if __has_builtin(__builtin_amdgcn_s_wait_asynccnt)
    __builtin_amdgcn_s_wait_asynccnt(0);
#else
    asm volatile("s_wait_asynccnt 0" ::: "memory");
#endif
    asm volatile("" ::: "memory");   // keep LDS reads after the wait
    float rbuf[8];
#pragma unroll
    for (int i = 0; i < 8; ++i) rbuf[i] = lds_rand[wave * 256 + lane * 8 + i];
#else
    const float* rp = rand_food + gbase + lane * 8;
    v4f r0 = __builtin_nontemporal_load((const v4f*)rp);
    v4f r1 = __builtin_nontemporal_load((const v4f*)(rp + 4));
    float rbuf[8] = { r0.x, r0.y, r0.z, r0.w, r1.x, r1.y, r1.z, r1.w };
#endif

    float bv = -2.0f; int bi = 0;
#pragma unroll
    for (int i = 0; i < 8; ++i) {
        const float key = (cells[i] == 0) ? rbuf[i] : -1.0f;
        if (key > bv) { bv = key; bi = lane * 8 + i; }   // '>' keeps first index
    }
    red_step<0x041F>(bv, bi);   // xor 1
    red_step<0x081F>(bv, bi);   // xor 2
    red_step<0x101F>(bv, bi);   // xor 4
    red_step<0x201F>(bv, bi);   // xor 8
    red_step<0x401F>(bv, bi);   // xor 16  -> all 32 lanes hold the argmax
    const int food = __builtin_amdgcn_readfirstlane(bi);

    if (spawn && (lane == (food >> 3))) {
        const int sub = food & 7;
#pragma unroll
        for (int i = 0; i < 8; ++i) if (sub == i) cells[i] = -1;
    }

    // ---- decrement body counters (suppressed while feeding) ----
    if (!feeding) {
#pragma unroll
        for (int i = 0; i < 8; ++i) if (cells[i] > 0) cells[i] -= 1;
    }

    // ---- new head: read at pos_cur AFTER decrement, write at pos_next last ----
    const int head = __builtin_amdgcn_readlane(sel8(cells, cur_lin & 7),
                                               cur_lin >> 3) + 1;
    if (lane == (next_lin >> 3)) {
        const int sub = next_lin & 7;
#pragma unroll
        for (int i = 0; i < 8; ++i) if (sub == i) cells[i] = head;
    }

    // ---- streaming store of the board as float (reference output dtype) ----
    float* op = out + gbase + lane * 8;
    v4f o0 = { (float)cells[0], (float)cells[1], (float)cells[2], (float)cells[3] };
    v4f o1 = { (float)cells[4], (float)cells[5], (float)cells[6], (float)cells[7] };
    __builtin_nontemporal_store(o0, (v4f*)op);
    __builtin_nontemporal_store(o1, (v4f*)(op + 4));
}

extern "C" void kernel_launch(void* const* d_in, const int* in_sizes, int n_in,
                              void* d_out, int out_size, void* d_ws, size_t ws_size,
                              hipStream_t stream) {
    const int*   action    = (const int*)  d_in[0];
    const int*   pos_prev  = (const int*)  d_in[1];
    const int*   pos_cur   = (const int*)  d_in[2];
    const int*   state     = (const int*)  d_in[3];
    const float* rand_food = (const float*)d_in[4];
    float*       out       = (float*)      d_out;

    const int G      = in_sizes[0];        // 524288 games
    const int blocks = (G + 7) / 8;        // 8 games (waves) per 256-thread block

    snake_step_kernel<<<blocks, 256, 0, stream>>>(action, pos_prev, pos_cur,
                                                  state, rand_food, out, G);
}